// QuantumLayer_45397804319444
// MI455X (gfx1250) — compile-verified
//
#include <hip/hip_runtime.h>
#include <math.h>

// ---------------------------------------------------------------------------
// Quantum layer on MI455X.
//  K1: angles = tanh(x @ W^T + b) * pi via V_WMMA_F32_16X16X4_F32.
//      W staged zero-padded in LDS -> no EXEC divergence in the K-loop.
//  K2: 14-qubit statevector sim, one WG per batch element, state (128 KB)
//      LDS-resident (WGP has 320 KB). Gates on wires 0..5 + CNOTs 0..4 are
//      fused into a register-resident pass (thread t owns idx = m*256+t,
//      m = bits 13..8): 18 LDS passes/layer instead of 28.
// ---------------------------------------------------------------------------

#define NQ      14
#define DIM     (1 << NQ)          // 16384 amplitudes
#define NLAYERS 4
#define BATCH   1024
#define TPB     256                // 8 wave32s
#define MREG    64                 // amplitudes per thread in registers

typedef __attribute__((ext_vector_type(2))) float v2f;
typedef __attribute__((ext_vector_type(8))) float v8f;

// ---------------------------------------------------------------------------
// Kernel 1: WMMA f32 16x16x4 pre-GEMM (N padded 14->16), one wave per tile.
// A (16x4): lanes 0-15 -> M=lane, {v0,v1}=K{0,1}; lanes 16-31 -> K{2,3}
// B (4x16): VGPR j, lanes 0-15 -> K=j, lanes 16-31 -> K=j+2; N = lane&15
// C (16x16): VGPR j -> M = j + 8*(lane>=16), N = lane&15
// ---------------------------------------------------------------------------
#define WPAD 514   // row stride (floats): even (b64-aligned) and !=0 mod 64
__global__ __launch_bounds__(32)
void ql_pre_gemm_wmma(const float* __restrict__ x,      // [1024][512]
                      const float* __restrict__ W,      // [14][512]
                      const float* __restrict__ bpre,   // [14]
                      float* __restrict__ angles)       // [1024][16] (ws)
{
    __shared__ float wl[16 * WPAD];                     // ~32 KB, rows 14/15 = 0
    const int lane = threadIdx.x & 31;

    for (int i = lane; i < 16 * 512; i += 32) {
        const int r = i >> 9, c = i & 511;
        wl[r * WPAD + c] = (r < NQ) ? W[r * 512 + c] : 0.0f;
    }
    __syncthreads();

    const int half = lane >> 4;        // K sub-pair select
    const int l16  = lane & 15;
    const int row0 = blockIdx.x * 16;
    const float* xrow = x + (size_t)(row0 + l16) * 512;
    const float* wrow = wl + l16 * WPAD;

    v8f acc = {};
    for (int k0 = 0; k0 < 512; k0 += 4) {
        const int ka = k0 + half * 2;
        v2f a, bfr;
        a.x   = xrow[ka];     a.y   = xrow[ka + 1];
        bfr.x = wrow[ka];     bfr.y = wrow[ka + 1];
        acc = __builtin_amdgcn_wmma_f32_16x16x4_f32(
                false, a, false, bfr, (short)0, acc, false, false);
    }

    const float PI_F = 3.14159265358979323846f;
    const float bp = (l16 < NQ) ? bpre[l16] : 0.0f;
#pragma unroll
    for (int j = 0; j < 8; ++j) {
        const int m = row0 + j + 8 * half;
        const float ang = tanhf(acc[j] + bp) * PI_F;
        if (l16 < NQ) angles[m * 16 + l16] = ang;
    }
}

// ---------------------------------------------------------------------------
// Kernel 2: statevector simulation.
// ---------------------------------------------------------------------------
__device__ __forceinline__ void rot_mat(const float* __restrict__ w,
                                        float& m00r, float& m00i, float& m01r, float& m01i,
                                        float& m10r, float& m10i, float& m11r, float& m11i)
{
    float s, c;   sincosf(0.5f * w[1], &s, &c);
    float sp, cp; sincosf(0.5f * (w[0] + w[2]), &sp, &cp); // ep = cp - i*sp
    float sm, cm; sincosf(0.5f * (w[0] - w[2]), &sm, &cm); // em = cm - i*sm
    m00r =  cp * c;  m00i = -sp * c;     // ep*c
    m01r = -cm * s;  m01i = -sm * s;     // -conj(em)*s
    m10r =  cm * s;  m10i = -sm * s;     // em*s
    m11r =  cp * c;  m11i =  sp * c;     // conj(ep)*c
}

__global__ __launch_bounds__(TPB)
void ql_state_sim(const float* __restrict__ angles,   // [1024][16]
                  const float* __restrict__ weights,  // [4][14][3]
                  float* __restrict__ out)            // [1024][14]
{
    __shared__ float sre[DIM];   // 64 KB
    __shared__ float sim_[DIM];  // 64 KB
    const int tid = threadIdx.x;
    const int b   = blockIdx.x;

    // ---- per-qubit initial 2-vectors: RY(th) RZ(th) H |0>
    float v0r[NQ], v0i[NQ], v1r[NQ], v1i[NQ];
    const float inv_s2 = 0.70710678118654752440f;
#pragma unroll
    for (int q = 0; q < NQ; ++q) {
        const float h = 0.5f * angles[b * 16 + q];
        float s, c;
        sincosf(h, &s, &c);
        const float ar = c * inv_s2, ai = -s * inv_s2;   // exp(-ih)/sqrt2
        const float br = c * inv_s2, bi =  s * inv_s2;   // exp(+ih)/sqrt2
        v0r[q] = c * ar - s * br;  v0i[q] = c * ai - s * bi;
        v1r[q] = s * ar + c * br;  v1i[q] = s * ai + c * bi;
    }

    // ---- initial product state directly in registers.
    // Thread owns idx = m*256 + tid. Wires 6..13 <-> idx bits 7..0 (= tid):
    // their contribution is one per-thread complex constant.
    float rr[MREG], ri[MREG];
    {
        float br_ = 1.0f, bi_ = 0.0f;
#pragma unroll
        for (int q = 6; q < NQ; ++q) {
            const int bit = (tid >> (NQ - 1 - q)) & 1;
            const float wr = bit ? v1r[q] : v0r[q];
            const float wi = bit ? v1i[q] : v0i[q];
            const float nr = br_ * wr - bi_ * wi;
            const float ni = br_ * wi + bi_ * wr;
            br_ = nr; bi_ = ni;
        }
#pragma unroll
        for (int m = 0; m < MREG; ++m) {
            float pr = br_, pq = bi_;
#pragma unroll
            for (int q = 0; q < 6; ++q) {           // wires 0..5 <-> m bits 5-q
                const int bit = (m >> (5 - q)) & 1;
                const float wr = bit ? v1r[q] : v0r[q];
                const float wi = bit ? v1i[q] : v0i[q];
                const float nr = pr * wr - pq * wi;
                const float ni = pr * wi + pq * wr;
                pr = nr; pq = ni;
            }
            rr[m] = pr; ri[m] = pq;
        }
    }

    for (int l = 0; l < NLAYERS; ++l) {
        if (l > 0) {
#pragma unroll
            for (int m = 0; m < MREG; ++m) {
                rr[m] = sre[m * TPB + tid];
                ri[m] = sim_[m * TPB + tid];
            }
        }

        // ---- register-resident Rot gates, wires 0..5 (state bits 13..8)
#pragma unroll
        for (int q = 0; q < 6; ++q) {
            float m00r, m00i, m01r, m01i, m10r, m10i, m11r, m11i;
            rot_mat(weights + (l * NQ + q) * 3, m00r, m00i, m01r, m01i,
                                                m10r, m10i, m11r, m11i);
            const int pb  = 5 - q;           // register-space bit
            const int str = 1 << pb;
#pragma unroll
            for (int g = 0; g < MREG / 2; ++g) {
                const int lo = g & (str - 1);
                const int hi = g >> pb;
                const int i0 = (hi << (pb + 1)) | lo;
                const int i1 = i0 | str;
                const float a0r = rr[i0], a0i = ri[i0];
                const float a1r = rr[i1], a1i = ri[i1];
                rr[i0] = m00r*a0r - m00i*a0i + m01r*a1r - m01i*a1i;
                ri[i0] = m00r*a0i + m00i*a0r + m01r*a1i + m01i*a1r;
                rr[i1] = m10r*a0r - m10i*a0i + m11r*a1r - m11i*a1i;
                ri[i1] = m10r*a0i + m10i*a0r + m11r*a1i + m11i*a1r;
            }
        }

        // ---- register-resident CNOTs q=0..4 (first five of the chain;
        //      control bit cb=5-q, target bit tb=4-q in register space)
#pragma unroll
        for (int q = 0; q < 5; ++q) {
            const int cb = 5 - q, tb = 4 - q;
#pragma unroll
            for (int g = 0; g < MREG / 4; ++g) {
                int xi = ((g >> tb) << (tb + 1)) | (g & ((1 << tb) - 1));
                xi = ((xi >> cb) << (cb + 1)) | (xi & ((1 << cb) - 1));
                const int i = xi | (1 << cb);
                const int j = i  | (1 << tb);
                const float tr = rr[i], ti = ri[i];
                rr[i] = rr[j]; ri[i] = ri[j];
                rr[j] = tr;    ri[j] = ti;
            }
        }

        // ---- publish register block (conflict-free: stride-1 across lanes)
#pragma unroll
        for (int m = 0; m < MREG; ++m) {
            sre[m * TPB + tid] = rr[m];
            sim_[m * TPB + tid] = ri[m];
        }
        __syncthreads();

        // ---- LDS Rot gates, wires 6..13 (state bit p = 13-q in 7..0)
        for (int q = 6; q < NQ; ++q) {
            float m00r, m00i, m01r, m01i, m10r, m10i, m11r, m11i;
            rot_mat(weights + (l * NQ + q) * 3, m00r, m00i, m01r, m01i,
                                                m10r, m10i, m11r, m11i);
            const int p      = NQ - 1 - q;
            const int stride = 1 << p;
            for (int t = 0; t < DIM / 2 / TPB; ++t) {
                const int g  = t * TPB + tid;
                const int lo = g & (stride - 1);
                const int hi = g >> p;
                const int i0 = (hi << (p + 1)) | lo;
                const int i1 = i0 | stride;
                const float a0r = sre[i0], a0i = sim_[i0];
                const float a1r = sre[i1], a1i = sim_[i1];
                sre[i0] = m00r*a0r - m00i*a0i + m01r*a1r - m01i*a1i;
                sim_[i0] = m00r*a0i + m00i*a0r + m01r*a1i + m01i*a1r;
                sre[i1] = m10r*a0r - m10i*a0i + m11r*a1r - m11i*a1i;
                sim_[i1] = m10r*a0i + m10i*a0r + m11r*a1i + m11i*a1r;
            }
            __syncthreads();
        }

        // ---- LDS CNOTs q=5..13 (in chain order)
        for (int q = 5; q < NQ; ++q) {
            const int cbit = NQ - 1 - q;
            const int tbit = NQ - 1 - ((q + 1) % NQ);
            const int blo = (cbit < tbit) ? cbit : tbit;
            const int bhi = (cbit < tbit) ? tbit : cbit;
            for (int t = 0; t < DIM / 4 / TPB; ++t) {
                const int g = t * TPB + tid;
                int xi = ((g >> blo) << (blo + 1)) | (g & ((1 << blo) - 1));
                xi = ((xi >> bhi) << (bhi + 1)) | (xi & ((1 << bhi) - 1));
                const int i = xi | (1 << cbit);       // control=1, target=0
                const int j = i  | (1 << tbit);
                const float tr = sre[i], ti = sim_[i];
                sre[i] = sre[j]; sim_[i] = sim_[j];
                sre[j] = tr;     sim_[j] = ti;
            }
            __syncthreads();
        }
    }

    // ---- <Z_q> = sum_i |amp_i|^2 * (1 - 2*bit(i, 13-q))
    float part[NQ];
#pragma unroll
    for (int q = 0; q < NQ; ++q) part[q] = 0.0f;
    for (int m = 0; m < MREG; ++m) {
        const int idx = m * TPB + tid;
        const float p = sre[idx] * sre[idx] + sim_[idx] * sim_[idx];
#pragma unroll
        for (int q = 0; q < NQ; ++q)
            part[q] += ((idx >> (NQ - 1 - q)) & 1) ? -p : p;
    }
    __syncthreads();   // state dead; reuse sre as reduction scratch

    const int laneId = tid & 31;
    const int waveId = tid >> 5;
#pragma unroll
    for (int q = 0; q < NQ; ++q) {
        float v = part[q];
        for (int off = 16; off > 0; off >>= 1)
            v += __shfl_down(v, off, 32);
        if (laneId == 0) sre[q * 8 + waveId] = v;
    }
    __syncthreads();
    if (tid < NQ) {
        float v = 0.0f;
#pragma unroll
        for (int w = 0; w < 8; ++w) v += sre[tid * 8 + w];
        out[b * NQ + tid] = v;
    }
}

// ---------------------------------------------------------------------------
extern "C" void kernel_launch(void* const* d_in, const int* in_sizes, int n_in,
                              void* d_out, int out_size, void* d_ws, size_t ws_size,
                              hipStream_t stream) {
    const float* x    = (const float*)d_in[0];  // [1024][512]
    const float* Wpre = (const float*)d_in[1];  // [14][512]
    const float* bpre = (const float*)d_in[2];  // [14]
    const float* wts  = (const float*)d_in[3];  // [4][14][3]
    float* out    = (float*)d_out;              // [1024][14]
    float* angles = (float*)d_ws;               // [1024][16] scratch

    ql_pre_gemm_wmma<<<BATCH / 16, 32, 0, stream>>>(x, Wpre, bpre, angles);
    ql_state_sim<<<BATCH, TPB, 0, stream>>>(angles, wts, out);
}